// GCN_81647328297445
// MI455X (gfx1250) — compile-verified
//
#include <hip/hip_runtime.h>
#include <hip/hip_bf16.h>

typedef __attribute__((ext_vector_type(2))) float v2f;
typedef __attribute__((ext_vector_type(8))) float v8f;

#define NNODES 100000
#define NEDGES 1600000
#define HDIM   64
#define NGRAPH 64

// ---------------- degree / normalization ----------------
__global__ void deg_kernel(const long long* __restrict__ dst, float* __restrict__ deg, int nedges) {
    int e = blockIdx.x * blockDim.x + threadIdx.x;
    if (e < nedges) atomicAdd(&deg[(int)dst[e]], 1.0f);
}

__global__ void dinv_kernel(const float* __restrict__ deg, float* __restrict__ dinv, int nnodes) {
    int n = blockIdx.x * blockDim.x + threadIdx.x;
    if (n < nnodes) dinv[n] = rsqrtf(deg[n] + 1.0f);
}

__global__ void coef_kernel(const long long* __restrict__ src, const long long* __restrict__ dst,
                            const float* __restrict__ dinv, float* __restrict__ coef, int nedges) {
    int e = blockIdx.x * blockDim.x + threadIdx.x;
    if (e < nedges) coef[e] = dinv[(int)src[e]] * dinv[(int)dst[e]];
}

// ---------------- WMMA matmul: T = in @ W ; O = T * self_coef ----------------
// block = 128 threads (4 waves). Each block: 64 rows x 64 cols of output.
// Wave w computes rows [row0+16w, row0+16w+16) x all 64 cols via 4 f32 16x16 tiles,
// accumulating with V_WMMA_F32_16X16X4_F32 over K=64 in steps of 4.
//
// W is staged K-pair-interleaved: sW[p*128 + 2n + q] = W[2p+q][n], so every B
// fragment (W[k][n], W[k+1][n]) is one aligned ds_load_b64 into an adjacent
// VGPR pair (no packing movs before the WMMA).
#define SA_STRIDE 66   // padded LDS row stride (floats) to avoid bank conflicts

__global__ void __launch_bounds__(128)
gcn_matmul(const float* __restrict__ in, const float* __restrict__ W,
           const float* __restrict__ dinv, float* __restrict__ T,
           float* __restrict__ O, int nrows) {
    __shared__ float sW[64 * 64];          // pair-interleaved layout
    __shared__ float sA[64 * SA_STRIDE];

    const int tid  = threadIdx.x;
    const int wave = tid >> 5;
    const int lane = tid & 31;
    const int row0 = blockIdx.x * 64;

    // Stage W (16 KB) into LDS in pair-interleaved layout.
    for (int i = tid; i < 64 * 16; i += 128) {
        int k  = i >> 4;            // W row 0..63
        int c4 = (i & 15) * 4;      // W col (float4 granularity)
        float4 v = ((const float4*)W)[i];
        float* base = sW + (k >> 1) * 128 + (k & 1);
        base[(c4 + 0) * 2] = v.x;
        base[(c4 + 1) * 2] = v.y;
        base[(c4 + 2) * 2] = v.z;
        base[(c4 + 3) * 2] = v.w;
    }
    // Stage 64 input rows into LDS (padded stride), float2, zero-fill OOB rows.
    for (int i = tid; i < 64 * 32; i += 128) {
        int r  = i >> 5;
        int c  = (i & 31) * 2;
        int gr = row0 + r;
        float2 v = make_float2(0.0f, 0.0f);
        if (gr < nrows) v = *(const float2*)(in + (size_t)gr * HDIM + c);
        *(float2*)(sA + r * SA_STRIDE + c) = v;
    }
    __syncthreads();

    v8f acc0 = {}, acc1 = {}, acc2 = {}, acc3 = {};
    const int mlocal = wave * 16 + (lane & 15);
    const int kk     = (lane >> 4) * 2;      // K sub-offset for this half-wave (even)
    const int ncol   = lane & 15;

    #pragma unroll
    for (int k = 0; k < HDIM; k += 4) {
        // A fragment: a[v] holds in[m][k+kk+v]
        v2f a = *(const v2f*)(sA + mlocal * SA_STRIDE + k + kk);
        // B fragments: contiguous (W[k+kk][n], W[k+kk+1][n]) pairs
        const float* bp = sW + ((k + kk) >> 1) * 128 + 2 * ncol;
        v2f b0 = *(const v2f*)(bp +  0);   // cols  0..15
        v2f b1 = *(const v2f*)(bp + 32);   // cols 16..31
        v2f b2 = *(const v2f*)(bp + 64);   // cols 32..47
        v2f b3 = *(const v2f*)(bp + 96);   // cols 48..63
        acc0 = __builtin_amdgcn_wmma_f32_16x16x4_f32(false, a, false, b0, (short)0, acc0, false, false);
        acc1 = __builtin_amdgcn_wmma_f32_16x16x4_f32(false, a, false, b1, (short)0, acc1, false, false);
        acc2 = __builtin_amdgcn_wmma_f32_16x16x4_f32(false, a, false, b2, (short)0, acc2, false, false);
        acc3 = __builtin_amdgcn_wmma_f32_16x16x4_f32(false, a, false, b3, (short)0, acc3, false, false);
    }

    // Epilogue: D layout -> row m = base + 8*(lane>=16) + v, col n = 16*t + (lane&15)
    const int gm_base = row0 + wave * 16 + (lane >> 4) * 8;
    v8f accs[4] = {acc0, acc1, acc2, acc3};

    if (row0 + 64 <= nrows) {
        // Fast path (all but the final partial block): branch-free.
        float di2[8];
        #pragma unroll
        for (int v = 0; v < 8; ++v) {
            float d = dinv[gm_base + v];
            di2[v] = d * d;
        }
        #pragma unroll
        for (int t = 0; t < 4; ++t) {
            #pragma unroll
            for (int v = 0; v < 8; ++v) {
                int idx = (gm_base + v) * HDIM + t * 16 + ncol;
                float val = accs[t][v];
                T[idx] = val;
                O[idx] = val * di2[v];
            }
        }
    } else {
        float di2[8];
        #pragma unroll
        for (int v = 0; v < 8; ++v) {
            int m = gm_base + v;
            float d = (m < nrows) ? dinv[m] : 0.0f;
            di2[v] = d * d;
        }
        #pragma unroll
        for (int t = 0; t < 4; ++t) {
            #pragma unroll
            for (int v = 0; v < 8; ++v) {
                int m = gm_base + v;
                if (m < nrows) {
                    int idx = m * HDIM + t * 16 + ncol;
                    float val = accs[t][v];
                    T[idx] = val;
                    O[idx] = val * di2[v];
                }
            }
        }
    }
}

// ---------------- edge scatter: O[dst] += T[src] * coef ----------------
// one wave32 per edge, 2 features per lane; edge metadata forced scalar (SMEM)
__global__ void gcn_edge(const long long* __restrict__ src, const long long* __restrict__ dst,
                         const float* __restrict__ coef, const float* __restrict__ T,
                         float* __restrict__ O, int nedges) {
    int wid = blockIdx.x * (blockDim.x >> 5) + (threadIdx.x >> 5);
    int e = __builtin_amdgcn_readfirstlane(wid);   // uniform -> scalar loads below
    if (e >= nedges) return;
    int lane = threadIdx.x & 31;
    int s = (int)src[e];
    int d = (int)dst[e];
    float c = coef[e];
    float2 hv = *(const float2*)(T + s * HDIM + lane * 2);
    atomicAdd(&O[d * HDIM + lane * 2 + 0], hv.x * c);
    atomicAdd(&O[d * HDIM + lane * 2 + 1], hv.y * c);
}

// ---------------- bias + relu (+ optional mean-pool accumulation) ----------------
__global__ void bias_relu_pool(float* __restrict__ O, const float* __restrict__ bias,
                               const long long* __restrict__ batch,
                               float* __restrict__ gsum, float* __restrict__ gcnt,
                               int nrows, int do_pool) {
    int t = blockIdx.x * blockDim.x + threadIdx.x;
    if (t >= nrows * HDIM) return;
    int n = t >> 6;
    int h = t & 63;
    float v = O[t] + bias[h];
    v = v > 0.0f ? v : 0.0f;
    O[t] = v;
    if (do_pool) {
        int g = (int)batch[n];
        atomicAdd(&gsum[g * HDIM + h], v);
        if (h == 0) atomicAdd(&gcnt[g], 1.0f);
    }
}

// ---------------- final head: out = (gsum/cnt) @ Wlin + blin ----------------
__global__ void final_head(const float* __restrict__ gsum, const float* __restrict__ gcnt,
                           const float* __restrict__ Wlin, const float* __restrict__ blin,
                           float* __restrict__ out) {
    int t = threadIdx.x;            // 128 = G*C
    int g = t >> 1, c = t & 1;
    float cnt = gcnt[g];
    cnt = cnt > 1.0f ? cnt : 1.0f;
    float inv = 1.0f / cnt;
    float sum = 0.0f;
    #pragma unroll
    for (int h = 0; h < HDIM; ++h)
        sum += gsum[g * HDIM + h] * inv * Wlin[h * 2 + c];
    out[g * 2 + c] = sum + blin[c];
}

extern "C" void kernel_launch(void* const* d_in, const int* in_sizes, int n_in,
                              void* d_out, int out_size, void* d_ws, size_t ws_size,
                              hipStream_t stream) {
    const float*     x     = (const float*)d_in[0];
    const long long* ei    = (const long long*)d_in[1];  // [2,E]: first E = src, next E = dst
    const long long* batch = (const long long*)d_in[2];
    const float* W0 = (const float*)d_in[3];  const float* b0 = (const float*)d_in[4];
    const float* W1 = (const float*)d_in[5];  const float* b1 = (const float*)d_in[6];
    const float* W2 = (const float*)d_in[7];  const float* b2 = (const float*)d_in[8];
    const float* Wl = (const float*)d_in[9];  const float* bl = (const float*)d_in[10];
    float* out = (float*)d_out;

    const int N = NNODES, E = NEDGES;
    const long long NH = (long long)N * HDIM;

    float* ws   = (float*)d_ws;
    float* deg  = ws;                 // N
    float* dinv = deg  + N;           // N
    float* coef = dinv + N;           // E
    float* bufA = coef + E;           // N*H  (raw matmul product T)
    float* bufB = bufA + NH;          // N*H
    float* bufC = bufB + NH;          // N*H
    float* gsum = bufC + NH;          // G*H
    float* gcnt = gsum + NGRAPH * HDIM; // G

    hipMemsetAsync(deg, 0, N * sizeof(float), stream);
    hipMemsetAsync(gsum, 0, (NGRAPH * HDIM + NGRAPH) * sizeof(float), stream);

    const long long* src = ei;
    const long long* dst = ei + E;

    deg_kernel <<<(E + 255) / 256, 256, 0, stream>>>(dst, deg, E);
    dinv_kernel<<<(N + 255) / 256, 256, 0, stream>>>(deg, dinv, N);
    coef_kernel<<<(E + 255) / 256, 256, 0, stream>>>(src, dst, dinv, coef, E);

    const int mblocks = (N + 63) / 64;
    const int eblocks = (E * 32 + 255) / 256;   // wave per edge
    const int nblocks = (N * HDIM + 255) / 256;

    // layer 1: in=x, T=bufA, O=bufB
    gcn_matmul<<<mblocks, 128, 0, stream>>>(x, W0, dinv, bufA, bufB, N);
    gcn_edge  <<<eblocks, 256, 0, stream>>>(src, dst, coef, bufA, bufB, E);
    bias_relu_pool<<<nblocks, 256, 0, stream>>>(bufB, b0, batch, gsum, gcnt, N, 0);

    // layer 2: in=bufB, T=bufA, O=bufC
    gcn_matmul<<<mblocks, 128, 0, stream>>>(bufB, W1, dinv, bufA, bufC, N);
    gcn_edge  <<<eblocks, 256, 0, stream>>>(src, dst, coef, bufA, bufC, E);
    bias_relu_pool<<<nblocks, 256, 0, stream>>>(bufC, b1, batch, gsum, gcnt, N, 0);

    // layer 3: in=bufC, T=bufA, O=bufB   (+ pooling)
    gcn_matmul<<<mblocks, 128, 0, stream>>>(bufC, W2, dinv, bufA, bufB, N);
    gcn_edge  <<<eblocks, 256, 0, stream>>>(src, dst, coef, bufA, bufB, E);
    bias_relu_pool<<<nblocks, 256, 0, stream>>>(bufB, b2, batch, gsum, gcnt, N, 1);

    final_head<<<1, 128, 0, stream>>>(gsum, gcnt, Wl, bl, out);
}